// GCNN_41188736369372
// MI455X (gfx1250) — compile-verified
//
#include <hip/hip_runtime.h>
#include <math.h>

typedef __attribute__((ext_vector_type(2))) float v2f;
typedef __attribute__((ext_vector_type(8))) float v8f;

#define LEAKY 0.01f

__device__ __forceinline__ void atomAdd(float* p, float v) {
  __hip_atomic_fetch_add(p, v, __ATOMIC_RELAXED, __HIP_MEMORY_SCOPE_AGENT);
}

// ---------------- degree / normalization ----------------

__global__ void k_deg_init(float* __restrict__ deg, int n) {
  int i = blockIdx.x * blockDim.x + threadIdx.x;
  if (i < n) deg[i] = 1.0f;  // self-loop weight contribution
}

__global__ void k_deg_accum(float* __restrict__ deg, const int* __restrict__ dst,
                            const float* __restrict__ w, int ne) {
  int e = blockIdx.x * blockDim.x + threadIdx.x;
  if (e < ne) atomAdd(&deg[dst[e]], w[e]);
}

__global__ void k_rsqrt(float* __restrict__ deg, int n) {
  int i = blockIdx.x * blockDim.x + threadIdx.x;
  if (i < n) {
    float d = deg[i];
    deg[i] = d > 0.0f ? rsqrtf(d) : 0.0f;  // deg now holds dis
  }
}

// ---------------- zero-pad weights to a multiple of 16 output rows ----------------
// Row-major [fout, fin] -> [foutPad, fin]; tail rows are exactly the tail elements.

__global__ void k_pad_w(const float* __restrict__ W, float* __restrict__ Wp,
                        int valid /* fout*fin */, int total /* foutPad*fin */) {
  int i = blockIdx.x * blockDim.x + threadIdx.x;
  if (i < total) Wp[i] = (i < valid) ? W[i] : 0.0f;
}

// ---------------- dense GEMM: H = X @ W^T  via V_WMMA_F32_16X16X4_F32 ----------------
// One wave per 16x16 output tile; FIN/FOUT compile-time so the K loop fully
// unrolls into a straight chain of global_load_b64 + v_wmma with no scalar
// loop overhead and no EXEC manipulation (Wp is padded; all loads unconditional).
// A (16x4 f32): lanes 0-15 hold K={k,k+1}, lanes 16-31 hold K={k+2,k+3}.
// B (4x16): same K split, N = lane&15.
// C/D: VGPR v -> M = v + 8*(lane>>4), N = lane&15.

template <int FIN, int FOUT>
__global__ __launch_bounds__(256)
void k_gemm_wmma(const float* __restrict__ X, const float* __restrict__ Wp,
                 float* __restrict__ H, int nrows) {
  constexpr int TILESN = (FOUT + 15) / 16;
  int lane = threadIdx.x & 31;
  int wave = blockIdx.x * (blockDim.x >> 5) + (threadIdx.x >> 5);
  int tileM = wave / TILESN;  // TILESN is 4/2/1 -> shifts
  int tileN = wave % TILESN;
  if (tileM * 16 >= nrows) return;  // whole-wave uniform exit

  int r = tileM * 16 + (lane & 15);
  int c = tileN * 16 + (lane & 15);
  int kh = (lane >> 4) << 1;  // 0 or 2
  const float* xp = X + (size_t)r * FIN + kh;
  const float* wp = Wp + (size_t)c * FIN + kh;

  v8f acc = {};
#pragma unroll
  for (int k = 0; k < FIN; k += 4) {
    v2f a, b;
    a.x = xp[k];
    a.y = xp[k + 1];
    b.x = wp[k];
    b.y = wp[k + 1];
    acc = __builtin_amdgcn_wmma_f32_16x16x4_f32(
        /*neg_a=*/false, a, /*neg_b=*/false, b,
        /*c_mod=*/(short)0, acc, /*reuse_a=*/false, /*reuse_b=*/false);
  }

  int mbase = tileM * 16 + ((lane >> 4) << 3);
  if ((FOUT % 16 == 0) || (c < FOUT)) {  // folds away for FOUT=64/32
#pragma unroll
    for (int v = 0; v < 8; ++v) {
      H[(size_t)(mbase + v) * FOUT + c] = acc[v];
    }
  }
}

// ---------------- edge aggregation: agg[dst] += dis[src]*w*dis[dst] * h[src] ----------------
// float4-vectorized: (fout/4) threads per edge; fout is a power of two.

__global__ __launch_bounds__(256)
void k_edge_agg(const float* __restrict__ h, float* __restrict__ agg,
                const int* __restrict__ src, const int* __restrict__ dst,
                const float* __restrict__ w, const float* __restrict__ dis,
                int ne, int f4shift, int fout) {
  unsigned gid = blockIdx.x * blockDim.x + threadIdx.x;
  unsigned total = (unsigned)ne << f4shift;
  if (gid >= total) return;
  unsigned e = gid >> f4shift;
  unsigned f4 = gid & ((1u << f4shift) - 1u);
  int s = src[e];
  int d = dst[e];
  float norm = dis[s] * w[e] * dis[d];
  float4 m = ((const float4*)(h + (size_t)s * fout))[f4];
  float* op = agg + (size_t)d * fout + (f4 << 2);
  atomAdd(op + 0, norm * m.x);
  atomAdd(op + 1, norm * m.y);
  atomAdd(op + 2, norm * m.z);
  atomAdd(op + 3, norm * m.w);
}

// ---------------- self-loop + bias + LeakyReLU ----------------

__global__ void k_finalize_leaky(const float* __restrict__ agg, const float* __restrict__ h,
                                 const float* __restrict__ dis, const float* __restrict__ b,
                                 float* __restrict__ xout, int total, int fshift) {
  int gid = blockIdx.x * blockDim.x + threadIdx.x;
  if (gid >= total) return;
  int i = gid >> fshift;
  int f = gid & ((1 << fshift) - 1);
  float di = dis[i];
  float v = agg[gid] + di * di * h[gid] + b[f];
  xout[gid] = v > 0.0f ? v : LEAKY * v;
}

// ---------------- self-loop + bias + softmax (fout == 4) ----------------

__global__ void k_finalize_softmax(const float* __restrict__ agg, const float* __restrict__ h,
                                   const float* __restrict__ dis, const float* __restrict__ b,
                                   float* __restrict__ out, int n) {
  int i = blockIdx.x * blockDim.x + threadIdx.x;
  if (i >= n) return;
  float di = dis[i];
  float d2 = di * di;
  float4 a = ((const float4*)agg)[i];
  float4 hh = ((const float4*)h)[i];
  float t0 = a.x + d2 * hh.x + b[0];
  float t1 = a.y + d2 * hh.y + b[1];
  float t2 = a.z + d2 * hh.z + b[2];
  float t3 = a.w + d2 * hh.w + b[3];
  float mx = fmaxf(fmaxf(t0, t1), fmaxf(t2, t3));
  float e0 = expf(t0 - mx);
  float e1 = expf(t1 - mx);
  float e2 = expf(t2 - mx);
  float e3 = expf(t3 - mx);
  float inv = 1.0f / (e0 + e1 + e2 + e3);
  float4 r;
  r.x = e0 * inv; r.y = e1 * inv; r.z = e2 * inv; r.w = e3 * inv;
  ((float4*)out)[i] = r;
}

// ---------------- host launcher ----------------

extern "C" void kernel_launch(void* const* d_in, const int* in_sizes, int n_in,
                              void* d_out, int out_size, void* d_ws, size_t ws_size,
                              hipStream_t stream) {
  const float* x  = (const float*)d_in[0];
  const int*   ei = (const int*)d_in[1];
  const float* w  = (const float*)d_in[2];
  const float* W1 = (const float*)d_in[3];
  const float* b1 = (const float*)d_in[4];
  const float* W2 = (const float*)d_in[5];
  const float* b2 = (const float*)d_in[6];
  const float* W3 = (const float*)d_in[7];
  const float* b3 = (const float*)d_in[8];
  float* out = (float*)d_out;

  const int N = in_sizes[0] / 128;   // 100000
  const int E = in_sizes[1] / 2;     // 3200000
  const int* src = ei;
  const int* dst = ei + E;

  // workspace carve-out
  char* ws = (char*)d_ws;
  size_t off = 0;
  auto carve = [&](size_t bytes) -> float* {
    float* p = (float*)(ws + off);
    off = (off + bytes + 255) & ~(size_t)255;
    return p;
  };
  float* dis  = carve((size_t)N * 4);        // deg -> dis (in place)
  float* h    = carve((size_t)N * 64 * 4);   // pre-aggregation features
  float* agg  = carve((size_t)N * 64 * 4);   // scatter-add target
  float* xcur = carve((size_t)N * 64 * 4);   // post-activation features
  float* Wp3  = carve((size_t)16 * 32 * 4);  // W3 zero-padded to 16 rows

  const int B = 256;

  // normalization: deg = 1 (self loop) + segment_sum(w, dst); dis = rsqrt(deg)
  k_deg_init<<<(N + B - 1) / B, B, 0, stream>>>(dis, N);
  k_deg_accum<<<(E + B - 1) / B, B, 0, stream>>>(dis, dst, w, E);
  k_rsqrt<<<(N + B - 1) / B, B, 0, stream>>>(dis, N);

  // pad W3 [4,32] -> [16,32] with zero rows
  k_pad_w<<<(16 * 32 + B - 1) / B, B, 0, stream>>>(W3, Wp3, 4 * 32, 16 * 32);

  // ---- layer 1: 128 -> 64 ----
  {
    const int fout = 64, tilesN = 4;
    int waves = (N / 16) * tilesN;
    k_gemm_wmma<128, 64><<<(waves + 7) / 8, 256, 0, stream>>>(x, W1, h, N);
    hipMemsetAsync(agg, 0, (size_t)N * fout * 4, stream);
    unsigned total = (unsigned)E << 4;  // fout/4 = 16 threads per edge
    k_edge_agg<<<(total + B - 1) / B, B, 0, stream>>>(h, agg, src, dst, w, dis, E, 4, fout);
    int t2 = N * fout;
    k_finalize_leaky<<<(t2 + B - 1) / B, B, 0, stream>>>(agg, h, dis, b1, xcur, t2, 6);
  }

  // ---- layer 2: 64 -> 32 ----
  {
    const int fout = 32, tilesN = 2;
    int waves = (N / 16) * tilesN;
    k_gemm_wmma<64, 32><<<(waves + 7) / 8, 256, 0, stream>>>(xcur, W2, h, N);
    hipMemsetAsync(agg, 0, (size_t)N * fout * 4, stream);
    unsigned total = (unsigned)E << 3;  // 8 threads per edge
    k_edge_agg<<<(total + B - 1) / B, B, 0, stream>>>(h, agg, src, dst, w, dis, E, 3, fout);
    int t2 = N * fout;
    k_finalize_leaky<<<(t2 + B - 1) / B, B, 0, stream>>>(agg, h, dis, b2, xcur, t2, 5);
  }

  // ---- layer 3: 32 -> 4, then softmax ----
  {
    const int fout = 4, tilesN = 1;
    int waves = (N / 16) * tilesN;
    k_gemm_wmma<32, 4><<<(waves + 7) / 8, 256, 0, stream>>>(xcur, Wp3, h, N);
    hipMemsetAsync(agg, 0, (size_t)N * fout * 4, stream);
    unsigned total = (unsigned)E;  // 1 thread per edge (float4)
    k_edge_agg<<<(total + B - 1) / B, B, 0, stream>>>(h, agg, src, dst, w, dis, E, 0, fout);
    k_finalize_softmax<<<(N + B - 1) / B, B, 0, stream>>>(agg, h, dis, b3, out, N);
  }
}